// CTCLayer_4492535792386
// MI455X (gfx1250) — compile-verified
//
#include <hip/hip_runtime.h>
#include <stdint.h>

// ---------------------------------------------------------------------------
// CTC forward loss (faithful port of K.ctc_batch_cost with full lengths).
// B=512, T=512, C=128 (blank=127), L=48, S=2L+1=97.
//
// MI455X reasoning: dominant cost is streaming y_pred once (134 MB -> ~5.8us
// at 23.3 TB/s). The t-recurrence is serial per batch row, so: one wave32 per
// row (512 independent waves), 4 consecutive extended states per lane (only
// ONE __shfl_up per step), and CDNA5 async global->LDS staging (ASYNCcnt,
// DEPTH=4 rows in flight) to hide memory latency behind the serial chain.
// WMMA does not apply: banded log-semiring recurrence, no dense matmul.
// ---------------------------------------------------------------------------

#define T_STEPS   512
#define C_CLASSES 128
#define L_LAB     48
#define WPB       8          // waves per block
#define DEPTH     4          // rows in flight per wave (power of two)
#define NEGF      (-1e30f)
#define EPSF      (1e-7f)

#if defined(__has_builtin)
#if __has_builtin(__builtin_amdgcn_global_load_async_to_lds_b128)
#define HAVE_ASYNC_LDS 1
#endif
#endif

#if defined(HAVE_ASYNC_LDS)
#pragma message("gfx1250: using __builtin_amdgcn_global_load_async_to_lds_b128 (ASYNCcnt path)")
#else
#pragma message("gfx1250: async-to-LDS builtin NOT available; using sync vmem+ds fallback")
#endif

// Builtin signature (from clang diagnostic): param0 is
//   '__attribute__((vector_size(16))) int __device__ *'  == int4 in AS1.
// So: (int4 AS1* gsrc, int4 AS3* lds_dst, imm offset, imm cpol).
typedef int v4i __attribute__((vector_size(16)));
typedef v4i __attribute__((address_space(1)))* g_v4i_p;   // global (AS1)
typedef v4i __attribute__((address_space(3)))* l_v4i_p;   // LDS (AS3, 32-bit)

#if defined(HAVE_ASYNC_LDS)
#if __has_builtin(__builtin_amdgcn_s_wait_asynccnt)
#define WAIT_ROWS() do { __builtin_amdgcn_s_wait_asynccnt(DEPTH - 1); \
                         asm volatile("" ::: "memory"); } while (0)
#else
#define WAIT_ROWS() asm volatile("s_wait_asynccnt %0" :: "n"(DEPTH - 1) : "memory")
#endif
#define PRE_STAGE_FENCE() asm volatile("s_wait_dscnt 0" ::: "memory")
#else
#define WAIT_ROWS() do { __builtin_amdgcn_wave_barrier(); \
                         asm volatile("" ::: "memory"); } while (0)
#define PRE_STAGE_FENCE() asm volatile("" ::: "memory")
#endif

// logaddexp(x, y); exact for one side == NEGF (finite "log zero").
__device__ __forceinline__ float lse2(float x, float y) {
    float m = fmaxf(x, y);
    float d = fabsf(x - y);
    return m + log1pf(__expf(-d));
}

// Stage 16 bytes per lane (a full 512B row per wave) global -> LDS.
__device__ __forceinline__ void stage_row(const float* gsrc, float* ldst) {
#if defined(HAVE_ASYNC_LDS)
    // Generic global pointer value == AS1 value: integer round-trip (no cast
    // between address spaces needed).
    g_v4i_p g = (g_v4i_p)(uintptr_t)gsrc;
    // Generic LDS pointer low 32 bits == LDS byte offset (aperture in [63:32]).
    l_v4i_p l = (l_v4i_p)(uint32_t)(uintptr_t)ldst;
    __builtin_amdgcn_global_load_async_to_lds_b128(g, l, 0, 0);
#else
    float4 v = *(const float4*)gsrc;
    *(float4*)ldst = v;
#endif
}

__global__ __launch_bounds__(WPB * 32) void ctc_forward_kernel(
        const int* __restrict__ y_true,     // [B, L]
        const float* __restrict__ y_pred,   // [B, T, C] softmax probs
        float* __restrict__ out)            // [B] (= [B,1] flattened)
{
    __shared__ __align__(16) float smem[WPB * DEPTH * C_CLASSES];

    const int lane = threadIdx.x & 31;
    const int wave = threadIdx.x >> 5;
    const int b    = blockIdx.x * WPB + wave;

    float* slots = &smem[wave * (DEPTH * C_CLASSES)];
    const float* rowbase = y_pred + (size_t)b * (T_STEPS * C_CLASSES);

    // ---- loop-invariant per-lane state: extended states s = 4*lane + r ----
    // r=0,2 -> blank(127); r=1 -> label 2*lane; r=3 -> label 2*lane+1.
    const bool lab_valid = (lane < (L_LAB / 2));   // lanes 0..23 hold real labels
    const int  lab0 = lab_valid ? y_true[b * L_LAB + 2 * lane]     : (C_CLASSES - 1);
    const int  lab1 = lab_valid ? y_true[b * L_LAB + 2 * lane + 1] : (C_CLASSES - 1);
    const int  labp = __shfl_up(lab1, 1);          // label 2*lane-1 (prev lane)
    const bool allow2_1 = lab_valid && ((lane == 0) || (lab0 != labp));
    const bool allow2_3 = lab_valid && (lab1 != lab0);

    // ---- prologue: rows 0..DEPTH-1 in flight ----
    for (int i = 0; i < DEPTH; ++i)
        stage_row(rowbase + i * C_CLASSES + lane * 4,
                  slots   + i * C_CLASSES + lane * 4);

    WAIT_ROWS();  // ASYNCcnt <= DEPTH-1  =>  row 0 resident (in-order returns)

    // ---- t = 0: alpha init (alpha[0]=lp_blank, alpha[1]=lp_label0) ----
    float a0, a1, a2, a3;
    {
        const float* row = slots;  // slot 0
        float lpB = __logf(row[C_CLASSES - 1] + EPSF);
        float lp1 = __logf(row[lab0] + EPSF);
        a0 = (lane == 0) ? lpB : NEGF;
        a1 = (lane == 0) ? lp1 : NEGF;
        a2 = NEGF;
        a3 = NEGF;
    }

    // ---- forward recurrence over t = 1..T-1 ----
    for (int t = 1; t < T_STEPS; ++t) {
        // Recycle slot (t-1)%DEPTH: its gather (ds_load) completed last iter;
        // fence DScnt before letting the async engine overwrite it.
        PRE_STAGE_FENCE();
        int tpre = t + (DEPTH - 1);
        if (tpre > T_STEPS - 1) tpre = T_STEPS - 1;   // clamp: keep ASYNCcnt uniform
        stage_row(rowbase + (size_t)tpre * C_CLASSES + lane * 4,
                  slots + ((t - 1) & (DEPTH - 1)) * C_CLASSES + lane * 4);

        WAIT_ROWS();  // exactly DEPTH in flight => row t resident
        const float* row = slots + (t & (DEPTH - 1)) * C_CLASSES;
        float lpB = __logf(row[C_CLASSES - 1] + EPSF);   // blank (r=0 and r=2)
        float lp1 = __logf(row[lab0] + EPSF);            // r=1
        float lp3 = __logf(row[lab1] + EPSF);            // r=3

        float p3 = __shfl_up(a3, 1);                     // alpha[s-1] for r=0
        if (lane == 0) p3 = NEGF;

        float n0 = lse2(a0, p3) + lpB;                   // s even: no skip
        float m1 = lse2(a1, a0);
        if (allow2_1) m1 = lse2(m1, p3);                 // skip from s-2 = prev a3
        float n1 = m1 + lp1;
        float n2 = lse2(a2, a1) + lpB;
        float m3 = lse2(a3, a2);
        if (allow2_3) m3 = lse2(m3, a1);                 // skip from s-2 = own a1
        float n3 = m3 + lp3;

        a0 = n0; a1 = n1; a2 = n2; a3 = n3;
    }

    // ---- loss = -logaddexp(alpha[96], alpha[95]) ----
    float v96 = __shfl(a0, 24);   // s=96 = 4*24+0
    float v95 = __shfl(a3, 23);   // s=95 = 4*23+3
    if (lane == 0) out[b] = -lse2(v95, v96);
}

extern "C" void kernel_launch(void* const* d_in, const int* in_sizes, int n_in,
                              void* d_out, int out_size, void* d_ws, size_t ws_size,
                              hipStream_t stream) {
    const int*   y_true = (const int*)d_in[0];    // [512, 48] int32
    const float* y_pred = (const float*)d_in[1];  // [512, 512, 128] float32
    float*       out    = (float*)d_out;          // [512, 1] float32
    (void)in_sizes; (void)n_in; (void)out_size; (void)d_ws; (void)ws_size;

    const int B = 512;
    dim3 grid(B / WPB);        // 64 blocks
    dim3 block(WPB * 32);      // 8 wave32 waves
    ctc_forward_kernel<<<grid, block, 0, stream>>>(y_true, y_pred, out);
}